// CellList_62388694942380
// MI455X (gfx1250) — compile-verified
//
#include <hip/hip_runtime.h>

// CellList masked-distance map on MI455X (gfx1250, wave32).
//
// out[i][j] = |r_i - r_j|  if (i > j && dist <= cutoff && sq > 0 && both species != -1) else 0
//
// Strategy: per 16x16 tile, one wave computes sq[i][j] = |ri|^2 + |rj|^2 - 2 ri.rj
// with a single V_WMMA_F32_16X16X4_F32 (A = [-2x,-2y,-2z,|ri|^2], B = [x,y,z,1],
// C = |rj|^2 broadcast). The WMMA value is a conservative screen; the rare
// passing elements (~0.8%) are recomputed in exact diff-form to match the
// reference bit-for-bit at the cutoff boundary. Dummy atoms poison norms with
// +inf so the screen rejects them. Kernel is store-BW bound (151 MB output).

#define N_ATOMS 6144
#define TILES   384            // N_ATOMS / 16
#define WAVES_PER_BLOCK 8      // 256 threads = 8 wave32

typedef __attribute__((ext_vector_type(2))) float v2f;
typedef __attribute__((ext_vector_type(8))) float v8f;

__global__ __launch_bounds__(256) void celllist_wmma_kernel(
    const int*   __restrict__ species,
    const float* __restrict__ coords,
    const int*   __restrict__ cutoffP,
    float*       __restrict__ out)
{
    const int lane = threadIdx.x & 31;
    const int wave = threadIdx.x >> 5;
    const int tile = blockIdx.x * WAVES_PER_BLOCK + wave;
    const int ti   = tile / TILES;
    const int tj   = tile - ti * TILES;

    const int  li = lane & 15;
    const bool hi = lane >= 16;

    if (ti < tj) {
        // Strictly upper tile: i < j everywhere -> all masked. Coalesced zero-fill:
        // each lane writes 32 contiguous bytes (2 x b128), 16 rows x 64B covered.
        const int r = lane >> 1;
        const int h = lane & 1;
        float4 z = {0.f, 0.f, 0.f, 0.f};
        float* p = out + (size_t)(ti * 16 + r) * N_ATOMS + (size_t)(tj * 16 + 8 * h);
        reinterpret_cast<float4*>(p)[0] = z;
        reinterpret_cast<float4*>(p)[1] = z;
        return;
    }

    // ---- compute tile (wave-uniform path from here: EXEC all ones for WMMA) ----
    const float cut    = (float)cutoffP[0];   // python int scalar -> int32[1]
    const float cut2   = cut * cut;
    const float screen = cut2 + 1.0f;         // WMMA sq abs-error << 1 for |coords|<=40

    const int rowA = ti * 16 + li;            // A-matrix row owned by this lane
    const int colB = tj * 16 + li;            // B/C/D column owned by this lane

    const float px = coords[3 * rowA + 0];
    const float py = coords[3 * rowA + 1];
    const float pz = coords[3 * rowA + 2];
    const float qx = coords[3 * colB + 0];
    const float qy = coords[3 * colB + 1];
    const float qz = coords[3 * colB + 2];

    const float INF = __builtin_inff();
    const float ni = (species[rowA] == -1) ? INF : (px * px + py * py + pz * pz);
    const float nj = (species[colB] == -1) ? INF : (qx * qx + qy * qy + qz * qz);

    // f32 16x4 A layout: lanes 0-15 -> (K0,K1), lanes 16-31 -> (K2,K3), M = lane&15.
    // A row = [-2x, -2y, -2z, ni];  B col = [x', y', z', 1];  C = nj (per column).
    v2f a, b;
    a.x = hi ? -2.0f * pz : -2.0f * px;
    a.y = hi ?  ni        : -2.0f * py;
    b.x = hi ?  qz        :  qx;
    b.y = hi ?  1.0f      :  qy;

    v8f c;
#pragma unroll
    for (int v = 0; v < 8; ++v) c[v] = nj;

    // D[m][n] = ni[m] + nj[n] - 2 * dot(r_m, r_n)  ==  |r_m - r_n|^2  (approx)
    v8f d = __builtin_amdgcn_wmma_f32_16x16x4_f32(
        /*neg_a=*/false, a, /*neg_b=*/false, b,
        /*c_mod=*/(short)0, c, /*reuse_a=*/false, /*reuse_b=*/false);

    // D layout: VGPR v -> M = v + (hi ? 8 : 0), N = lane&15.
    const int m0 = ti * 16 + (hi ? 8 : 0);
#pragma unroll
    for (int v = 0; v < 8; ++v) {
        const int m  = m0 + v;
        float val = 0.0f;
        // Conservative screen (NaN/Inf -> false -> masked 0). Exact refine for
        // the ~0.8% in-cutoff elements: same math as the reference.
        if (m > colB && d[v] <= screen) {
            const float dx = coords[3 * m + 0] - qx;   // L1-broadcast hits
            const float dy = coords[3 * m + 1] - qy;
            const float dz = coords[3 * m + 2] - qz;
            const float sq = dx * dx + dy * dy + dz * dz;
            const float dist = __builtin_sqrtf(sq > 0.0f ? sq : 1.0f);
            if (sq > 0.0f && dist <= cut) val = dist;
        }
        out[(size_t)m * N_ATOMS + (size_t)colB] = val;
    }
}

extern "C" void kernel_launch(void* const* d_in, const int* in_sizes, int n_in,
                              void* d_out, int out_size, void* d_ws, size_t ws_size,
                              hipStream_t stream) {
    const int*   species = (const int*)d_in[0];   // [1, 6144] int32
    const float* coords  = (const float*)d_in[1]; // [1, 6144, 3] float32
    const int*   cutoff  = (const int*)d_in[2];   // scalar int
    float*       out     = (float*)d_out;         // [6144, 6144] float32

    const int nTiles = TILES * TILES;             // 147456
    const int blocks = nTiles / WAVES_PER_BLOCK;  // 18432
    celllist_wmma_kernel<<<blocks, 256, 0, stream>>>(species, coords, cutoff, out);
}